// PlmConvKnrm_82463372083432
// MI455X (gfx1250) — compile-verified
//
#include <hip/hip_runtime.h>
#include <hip/hip_bf16.h>
#include <math.h>

typedef __attribute__((ext_vector_type(2))) float v2f;
typedef __attribute__((ext_vector_type(8))) float v8f;

// ---------------- problem constants ----------------
constexpr int BN = 64, HH = 768, FF = 128, MAXLEN = 512;
// conv output lengths
constexpr int LQ0 = 31, LQ1 = 30, LQ2 = 29;      // query: 31 tokens, VALID conv
constexpr int LD0 = 477, LD1 = 476, LD2 = 475;   // doc: 477 tokens, VALID conv

// ---------------- workspace layout (float offsets) ----------------
constexpr size_t OFF_WR0 = 0;
constexpr size_t OFF_WR1 = OFF_WR0 + (size_t)768 * FF;
constexpr size_t OFF_WR2 = OFF_WR1 + (size_t)1536 * FF;
constexpr size_t OFF_QC0 = OFF_WR2 + (size_t)2304 * FF;
constexpr size_t OFF_QC1 = OFF_QC0 + (size_t)BN * LQ0 * FF;
constexpr size_t OFF_QC2 = OFF_QC1 + (size_t)BN * LQ1 * FF;
constexpr size_t OFF_DC0 = OFF_QC2 + (size_t)BN * LQ2 * FF;
constexpr size_t OFF_DC1 = OFF_DC0 + (size_t)BN * LD0 * FF;
constexpr size_t OFF_DC2 = OFF_DC1 + (size_t)BN * LD1 * FF;
constexpr size_t OFF_IQ0 = OFF_DC2 + (size_t)BN * LD2 * FF;
constexpr size_t OFF_IQ1 = OFF_IQ0 + (size_t)BN * LQ0;
constexpr size_t OFF_IQ2 = OFF_IQ1 + (size_t)BN * LQ1;
constexpr size_t OFF_ID0 = OFF_IQ2 + (size_t)BN * LQ2;
constexpr size_t OFF_ID1 = OFF_ID0 + (size_t)BN * LD0;
constexpr size_t OFF_ID2 = OFF_ID1 + (size_t)BN * LD1;
constexpr size_t OFF_FEATS = OFF_ID2 + (size_t)BN * LD2;
constexpr size_t WS_FLOATS = OFF_FEATS + (size_t)BN * 99;

// RBF constants: means, 1/(2*sigma^2)  (sigma = 0.001, then 0.1 x10)
constexpr float C_MEANS[11] = {1.0f, 0.9f, 0.7f, 0.5f, 0.3f, 0.1f,
                               -0.1f, -0.3f, -0.5f, -0.7f, -0.9f};
constexpr float C_I2V[11] = {500000.0f, 50.0f, 50.0f, 50.0f, 50.0f, 50.0f,
                             50.0f, 50.0f, 50.0f, 50.0f, 50.0f};

// ---------------- kernels ----------------

__global__ void zero_feats_kernel(float* ws) {
  int idx = blockIdx.x * blockDim.x + threadIdx.x;
  if (idx < BN * 99) ws[OFF_FEATS + idx] = 0.0f;
}

// Transposed rearrange: Wrt[f*Kt + k] = W[f][h][t] with k = t*768 + h.
// (F,K) layout -> the conv GEMM's B operand becomes one contiguous b64 per lane.
__global__ void rearrange_w_kernel(const float* __restrict__ W, float* __restrict__ ws,
                                   size_t wrOff, int w) {
  int idx = blockIdx.x * blockDim.x + threadIdx.x;
  int Kt = 768 * w;
  int total = Kt * 128;
  if (idx < total) {
    int f = idx / Kt;
    int k = idx - f * Kt;
    int t = k / 768;
    int h = k - t * 768;
    ws[wrOff + idx] = W[(size_t)f * 768 * w + (size_t)h * w + t];
  }
}

// Conv-as-GEMM on the f32 WMMA pipe.  One wave per 16x16 output tile.
// A row p = X_flat[b, (posOff+p)*768 : ... + Ktot]  (sliding rows, stride 768)
// B column f = Wrt[f, :]  (contiguous K, so each lane issues one b64 per WMMA)
__global__ __launch_bounds__(32) void conv_gemm_kernel(
    const float* __restrict__ X, float* __restrict__ ws,
    const float* __restrict__ bias, size_t wrOff, size_t outOff,
    int P, int Ktot, int posOff) {
  const int lane = threadIdx.x;
  const int half = lane >> 4;   // K-pair selector for the A/B wave32 layouts
  const int l = lane & 15;
  const int b = blockIdx.x;
  const int m0 = blockIdx.y * 16;
  const int n0 = blockIdx.z * 16;

  int ra = m0 + l;
  if (ra > P - 1) ra = P - 1;  // clamp: reads stay in-bounds, results discarded
  const float2* xrow2 =
      (const float2*)(X + (size_t)b * MAXLEN * HH + (size_t)(posOff + ra) * HH);
  const int col = n0 + l;
  const float2* wcol2 = (const float2*)(ws + wrOff + (size_t)col * Ktot);

  const int Ksteps = Ktot >> 2;
  v8f c = {};
#pragma unroll 8
  for (int k = 0; k < Ksteps; ++k) {  // each step covers K=4
    float2 t = xrow2[2 * k + half];   // A[m=l][4k + 2*half (+1)]
    v2f a;  a.x = t.x;  a.y = t.y;
    float2 u = wcol2[2 * k + half];   // B[4k + 2*half (+1)][n=l]
    v2f bb; bb.x = u.x; bb.y = u.y;
    c = __builtin_amdgcn_wmma_f32_16x16x4_f32(false, a, false, bb, (short)0, c,
                                              false, false);
  }

  float bv = bias[col];
  float* outp = ws + outOff + (size_t)b * P * FF + col;
#pragma unroll
  for (int r = 0; r < 8; ++r) {  // C: VGPR r -> row m0 + 8*half + r, col = l
    int row = m0 + half * 8 + r;
    if (row < P) {
      float v = c[r] + bv;
      outp[(size_t)row * FF] = v > 0.0f ? v : 0.0f;
    }
  }
}

// inverse L2 norm of each 128-wide row; one wave per row
__global__ __launch_bounds__(32) void norm_kernel(const float* __restrict__ buf,
                                                  float* __restrict__ out) {
  int row = blockIdx.x;
  int lane = threadIdx.x;
  const float* p = buf + (size_t)row * FF + lane * 4;
  float s = p[0] * p[0] + p[1] * p[1] + p[2] * p[2] + p[3] * p[3];
  s += __shfl_xor(s, 16, 32);
  s += __shfl_xor(s, 8, 32);
  s += __shfl_xor(s, 4, 32);
  s += __shfl_xor(s, 2, 32);
  s += __shfl_xor(s, 1, 32);
  if (lane == 0) out[row] = 1.0f / sqrtf(s);
}

// sim (f32 WMMA) + masked cosine + RBF log-pool.  One wave per (b, pair, q-tile).
__global__ __launch_bounds__(32) void sim_rbf_kernel(float* __restrict__ ws,
                                                     const int* __restrict__ qlens,
                                                     const int* __restrict__ dlens) {
  const int lane = threadIdx.x;
  const int half = lane >> 4;
  const int l = lane & 15;
  const int b = blockIdx.x;
  const int pair = blockIdx.y;
  const int i = pair / 3, j = pair % 3;
  const int q0 = blockIdx.z * 16;

  const float* QC; const float* IQ; int Lq;
  if (i == 0)      { QC = ws + OFF_QC0; IQ = ws + OFF_IQ0; Lq = LQ0; }
  else if (i == 1) { QC = ws + OFF_QC1; IQ = ws + OFF_IQ1; Lq = LQ1; }
  else             { QC = ws + OFF_QC2; IQ = ws + OFF_IQ2; Lq = LQ2; }
  const float* DC; const float* ID; int Ld;
  if (j == 0)      { DC = ws + OFF_DC0; ID = ws + OFF_ID0; Ld = LD0; }
  else if (j == 1) { DC = ws + OFF_DC1; ID = ws + OFF_ID1; Ld = LD1; }
  else             { DC = ws + OFF_DC2; ID = ws + OFF_ID2; Ld = LD2; }

  QC += (size_t)b * Lq * FF;  IQ += (size_t)b * Lq;
  DC += (size_t)b * Ld * FF;  ID += (size_t)b * Ld;
  const int qlen = qlens[b], dlen = dlens[b];

  int qa = q0 + l; if (qa > Lq - 1) qa = Lq - 1;
  const float2* qrow2 = (const float2*)(QC + (size_t)qa * FF);

  // inv query norm per C-row; 0 when masked/out-of-range -> sim becomes 0 as in reference
  float invq[8];
#pragma unroll
  for (int r = 0; r < 8; ++r) {
    int q = q0 + half * 8 + r;
    invq[r] = (q < Lq && q <= qlen) ? IQ[q] : 0.0f;
  }

  float acc[8][11];
#pragma unroll
  for (int r = 0; r < 8; ++r)
#pragma unroll
    for (int m = 0; m < 11; ++m) acc[r][m] = 0.0f;

  for (int d0 = 0; d0 < 480; d0 += 16) {  // ceil(477/16)=30 tiles for all j
    int da = d0 + l; if (da > Ld - 1) da = Ld - 1;
    const float2* drow2 = (const float2*)(DC + (size_t)da * FF);

    v8f c = {};
#pragma unroll
    for (int k = 0; k < 32; ++k) {  // K=128, 4 per WMMA
      float2 t = qrow2[2 * k + half];
      v2f a;  a.x = t.x;  a.y = t.y;
      float2 u = drow2[2 * k + half];  // B[k][n=l] = D[d0+l][k]
      v2f bb; bb.x = u.x; bb.y = u.y;
      c = __builtin_amdgcn_wmma_f32_16x16x4_f32(false, a, false, bb, (short)0, c,
                                                false, false);
    }

    int dcol = d0 + l;
    bool dIn = dcol < Ld;                                  // column exists
    float invd = (dIn && dcol <= dlen) ? ID[dcol] : 0.0f;  // mask -> sim = 0
    if (dIn) {
#pragma unroll
      for (int r = 0; r < 8; ++r) {
        float s = c[r] * invq[r] * invd;
#pragma unroll
        for (int m = 0; m < 11; ++m) {
          float df = s - C_MEANS[m];
          acc[r][m] += __expf(-(df * df) * C_I2V[m]);
        }
      }
    }
  }

  // reduce the doc-dimension partials across the 16 lanes of each half-wave
#pragma unroll
  for (int r = 0; r < 8; ++r)
#pragma unroll
    for (int m = 0; m < 11; ++m) {
      float v = acc[r][m];
      v += __shfl_xor(v, 1, 32);
      v += __shfl_xor(v, 2, 32);
      v += __shfl_xor(v, 4, 32);
      v += __shfl_xor(v, 8, 32);
      acc[r][m] = v;
    }

  if (l == 0) {  // lane 0 (rows q0..q0+7) and lane 16 (rows q0+8..q0+15)
    float* feats = ws + OFF_FEATS + (size_t)b * 99 + (size_t)pair * 11;
#pragma unroll
    for (int m = 0; m < 11; ++m) {
      float t = 0.0f;
#pragma unroll
      for (int r = 0; r < 8; ++r) {
        int q = q0 + half * 8 + r;
        if (q < Lq) t += __logf(acc[r][m]);
      }
      atomicAdd(&feats[m], t);
    }
  }
}

__global__ void head_kernel(const float* __restrict__ ws,
                            const float* __restrict__ pw,
                            const float* __restrict__ pb,
                            float* __restrict__ out) {
  int b = threadIdx.x;
  if (b < BN) {
    const float* f = ws + OFF_FEATS + (size_t)b * 99;
    float s = pb[0];
    for (int m = 0; m < 99; ++m) s += f[m] * pw[m];
    out[b] = tanhf(s);
  }
}

// ---------------- launcher ----------------
extern "C" void kernel_launch(void* const* d_in, const int* in_sizes, int n_in,
                              void* d_out, int out_size, void* d_ws, size_t ws_size,
                              hipStream_t stream) {
  (void)in_sizes; (void)n_in; (void)out_size; (void)ws_size;
  const float* X = (const float*)d_in[0];
  const int* qlens = (const int*)d_in[1];
  const int* dlens = (const int*)d_in[2];
  const float* cw[3] = {(const float*)d_in[3], (const float*)d_in[5],
                        (const float*)d_in[7]};
  const float* cb[3] = {(const float*)d_in[4], (const float*)d_in[6],
                        (const float*)d_in[8]};
  const float* pw = (const float*)d_in[9];
  const float* pb = (const float*)d_in[10];
  float* ws = (float*)d_ws;
  float* out = (float*)d_out;

  zero_feats_kernel<<<(BN * 99 + 255) / 256, 256, 0, stream>>>(ws);

  const size_t wrOff[3] = {OFF_WR0, OFF_WR1, OFF_WR2};
  const size_t qcOff[3] = {OFF_QC0, OFF_QC1, OFF_QC2};
  const size_t dcOff[3] = {OFF_DC0, OFF_DC1, OFF_DC2};
  const size_t iqOff[3] = {OFF_IQ0, OFF_IQ1, OFF_IQ2};
  const size_t idOff[3] = {OFF_ID0, OFF_ID1, OFF_ID2};
  const int LQa[3] = {LQ0, LQ1, LQ2};
  const int LDa[3] = {LD0, LD1, LD2};

  for (int i = 0; i < 3; ++i) {
    int w = i + 1;
    int Ktot = 768 * w;
    int total = Ktot * 128;
    rearrange_w_kernel<<<(total + 255) / 256, 256, 0, stream>>>(cw[i], ws,
                                                                wrOff[i], w);
    dim3 gq(BN, (LQa[i] + 15) / 16, 8);
    conv_gemm_kernel<<<gq, 32, 0, stream>>>(X, ws, cb[i], wrOff[i], qcOff[i],
                                            LQa[i], Ktot, /*posOff=*/1);
    dim3 gd(BN, (LDa[i] + 15) / 16, 8);
    conv_gemm_kernel<<<gd, 32, 0, stream>>>(X, ws, cb[i], wrOff[i], dcOff[i],
                                            LDa[i], Ktot, /*posOff=*/33);
    norm_kernel<<<dim3(BN * LQa[i]), 32, 0, stream>>>(ws + qcOff[i], ws + iqOff[i]);
    norm_kernel<<<dim3(BN * LDa[i]), 32, 0, stream>>>(ws + dcOff[i], ws + idOff[i]);
  }

  sim_rbf_kernel<<<dim3(BN, 9, 2), 32, 0, stream>>>(ws, qlens, dlens);
  head_kernel<<<1, 64, 0, stream>>>(ws, pw, pb, out);
}